// HyperNetworkDecoder_24927990186109
// MI455X (gfx1250) — compile-verified
//
#include <hip/hip_runtime.h>
#include <math.h>

typedef __attribute__((ext_vector_type(2))) float v2f;
typedef __attribute__((ext_vector_type(8))) float v8f;

#define KDIM 2
#define PDIM 4096
#define WDIM 64
#define NIN0 126
#define NPAD0 128
#define RS 132                               // LDS row stride (floats), bank-conflict free
#define WSLAB (NPAD0*WDIM + 7*WDIM*WDIM)     // 36864 floats of Wt per k
#define WS_B_OFF (KDIM*WSLAB)                // bias table offset in workspace (floats)

// ---------------------------------------------------------------------------
// Kernel 1: hypernetwork weight generation.
// One block per (k, layer).  Produces, per (k,l), the layer weight matrix
// stored directly in WMMA B-fragment order:
//   for N-tile nt (0..3), k-step s (k0 = 4s), lane = hi*16 + col (col = N&15):
//     frag[((nt*STEPS + s)*32 + lane)*2 + r] = Wt[k0 + 2*hi + r][nt*16 + col]
//   where Wt[m][n] = (h @ wW + wb)[n*n_in + m]  (zero-padded rows m >= n_in)
// plus bias[n] = (h @ bbW + bbb)[n], with h = relu(LN(chunk @ hW + hb)).
// ---------------------------------------------------------------------------
__global__ __launch_bounds__(256) void hyper_weights(
    const float* __restrict__ lat,  const float* __restrict__ temb,
    const float* __restrict__ hW0,  const float* __restrict__ hb0,
    const float* __restrict__ hls0, const float* __restrict__ hlb0,
    const float* __restrict__ wW0,  const float* __restrict__ wb0,
    const float* __restrict__ bbW0, const float* __restrict__ bbb0,
    const float* __restrict__ hW,   const float* __restrict__ hb,
    const float* __restrict__ hls,  const float* __restrict__ hlb,
    const float* __restrict__ wW,   const float* __restrict__ wb,
    const float* __restrict__ bbW,  const float* __restrict__ bbb,
    float* __restrict__ wsW, float* __restrict__ wsB)
{
    const int blk = blockIdx.x;
    const int k   = blk >> 3;
    const int l   = blk & 7;
    const int tid = threadIdx.x;

    __shared__ float chunk[128];
    __shared__ float t[WDIM];
    __shared__ float h[WDIM];

    // chunk = [latent_tokens[k,l,:], token_embeddings[l,:]]
    if (tid < 128)
        chunk[tid] = (tid < 64) ? lat[(k*8 + l)*64 + tid]
                                : temb[l*64 + (tid - 64)];
    __syncthreads();

    const float* HW  = (l == 0) ? hW0  : hW  + (l-1)*128*WDIM;
    const float* HB  = (l == 0) ? hb0  : hb  + (l-1)*WDIM;
    const float* HLS = (l == 0) ? hls0 : hls + (l-1)*WDIM;
    const float* HLB = (l == 0) ? hlb0 : hlb + (l-1)*WDIM;

    if (tid < WDIM) {
        float acc = HB[tid];
        for (int j = 0; j < 128; ++j)
            acc = fmaf(chunk[j], HW[j*WDIM + tid], acc);
        t[tid] = acc;
    }
    __syncthreads();

    if (tid < WDIM) {
        float s = 0.f, ss = 0.f;
        for (int j = 0; j < WDIM; ++j) { float x = t[j]; s += x; ss += x*x; }
        float mu  = s  * (1.f/WDIM);
        float var = ss * (1.f/WDIM) - mu*mu;
        float x = (t[tid] - mu) * rsqrtf(var + 1e-6f) * HLS[tid] + HLB[tid];
        h[tid] = fmaxf(x, 0.f);
    }
    __syncthreads();

    const int n_in = (l == 0) ? NIN0  : WDIM;
    const int npad = (l == 0) ? NPAD0 : WDIM;
    const int steps = npad >> 2;                        // k-steps of 4
    const int cols = n_in * WDIM;                       // 8064 or 4096
    const float* WW = (l == 0) ? wW0 : wW + (l-1)*WDIM*4096;
    const float* WB = (l == 0) ? wb0 : wb + (l-1)*4096;
    float* outW = wsW + (k*WSLAB + ((l == 0) ? 0 : NPAD0*WDIM + (l-1)*WDIM*WDIM));

    for (int idx = tid; idx < npad*WDIM; idx += 256) {
        int m = idx >> 6;          // input-feature index (row of Wt)
        int n = idx & 63;          // output index
        float v = 0.f;
        if (m < n_in) {
            int j = n*n_in + m;    // flat index into (h @ wW) before reshape
            v = WB[j];
            for (int i = 0; i < WDIM; ++i)
                v = fmaf(h[i], WW[i*cols + j], v);
        }
        // scatter into B-fragment order
        int s   = m >> 2;
        int hi2 = (m >> 1) & 1;
        int r   = m & 1;
        int nt  = n >> 4;
        int col = n & 15;
        int fidx = ((nt*steps + s)*32 + hi2*16 + col)*2 + r;
        outW[fidx] = v;            // zero-padded rows m = 126,127 for layer 0
    }

    const float* BW = (l == 0) ? bbW0 : bbW + (l-1)*WDIM*WDIM;
    const float* BB = (l == 0) ? bbb0 : bbb + (l-1)*WDIM;
    if (tid < WDIM) {
        float v = BB[tid];
        for (int i = 0; i < WDIM; ++i)
            v = fmaf(h[i], BW[i*WDIM + tid], v);
        wsB[(k*8 + l)*WDIM + tid] = v;
    }
}

// ---------------------------------------------------------------------------
// One MLP layer for a 16-ray tile, executed by one wave32 using
// V_WMMA_F32_16X16X4_F32:   D(16x16) += A(16x4:net) x B(4x16:Wt-fragment)
// C is initialized with the broadcast bias (bias is constant along M, which
// matches the D layout: lane -> col, VGPR -> row).
// ---------------------------------------------------------------------------
template <int NPAD>
__device__ __forceinline__ void layer_step(
    float* __restrict__ net, int m, int hi, int lane,
    const float* __restrict__ Wt, const float* __restrict__ bias,
    const float* __restrict__ lnS, const float* __restrict__ lnB)
{
    constexpr int STEPS = NPAD >> 2;
    const v2f* __restrict__ Wf = (const v2f*)Wt;   // fragment-packed, 8B aligned

    v8f acc[4];
#pragma unroll
    for (int nt = 0; nt < 4; ++nt) {
        float bv = bias[nt*16 + m];                // bias rides the C operand
        v8f c = {bv, bv, bv, bv, bv, bv, bv, bv};
        for (int s = 0; s < STEPS; ++s) {
            int k0 = 4*s;
            v2f a, b;
            // A (16x4): lane = M | (hi<<4); VGPR0 holds K = 2*hi, VGPR1 holds K = 2*hi+1
            a.x = net[m*RS + k0 + 2*hi];
            a.y = net[m*RS + k0 + 2*hi + 1];
            // B (4x16): one coalesced b64 per lane from the fragment-packed table
            b = Wf[(nt*STEPS + s)*32 + lane];
            c = __builtin_amdgcn_wmma_f32_16x16x4_f32(
                    false, a, false, b, (short)0, c, false, false);
        }
        acc[nt] = c;
    }

    // D layout: VGPR v, lane -> row (v + 8*hi), col (nt*16 + m)
#pragma unroll
    for (int nt = 0; nt < 4; ++nt)
#pragma unroll
        for (int v = 0; v < 8; ++v)
            net[(v + 8*hi)*RS + nt*16 + m] = acc[nt][v];
    __syncthreads();

    // LayerNorm + ReLU (bias already in), one lane per ray (lanes 0..15)
    if (hi == 0) {
        float* row = net + m*RS;
        float s = 0.f, ss = 0.f;
        for (int n = 0; n < WDIM; ++n) { float x = row[n]; s += x; ss += x*x; }
        float mu  = s  * (1.f/WDIM);
        float var = ss * (1.f/WDIM) - mu*mu;
        float inv = rsqrtf(var + 1e-6f);
        for (int n = 0; n < WDIM; ++n) {
            float y = (row[n] - mu) * inv * lnS[n] + lnB[n];
            row[n] = fmaxf(y, 0.f);
        }
    }
    __syncthreads();
}

// ---------------------------------------------------------------------------
// Kernel 2: fused posenc + 8 hyper-layers + rgb head.
// 128 threads = 4 wave32; each wave owns a 16-ray tile and a private LDS slab.
// ---------------------------------------------------------------------------
__global__ __launch_bounds__(128) void mlp_forward(
    const float* __restrict__ rays,
    const float* __restrict__ wsW, const float* __restrict__ wsB,
    const float* __restrict__ ln_s0, const float* __restrict__ ln_b0,
    const float* __restrict__ ln_s,  const float* __restrict__ ln_b,
    const float* __restrict__ rgbW,  const float* __restrict__ rgbB,
    float* __restrict__ out)
{
    __shared__ float sNet[4 * 16 * RS];

    const int tid  = threadIdx.x;
    const int wave = tid >> 5;
    const int lane = tid & 31;
    const int hi   = lane >> 4;
    const int m    = lane & 15;

    float* net = sNet + wave * (16 * RS);

    const int rayBase = blockIdx.x * 64 + wave * 16;   // 64 rays/block, all in one k
    const int k       = rayBase >> 12;                 // 4096 rays per k
    const int gray    = rayBase + m;

    // ---- positional encoding: plucker + sin/cos bands ----
    {
        const float* r = rays + gray*6;
        float ox = r[0], oy = r[1], oz = r[2];
        float dx = r[3], dy = r[4], dz = r[5];
        float pl[6];
        pl[0] = dx; pl[1] = dy; pl[2] = dz;
        pl[3] = oy*dz - oz*dy;
        pl[4] = oz*dx - ox*dz;
        pl[5] = ox*dy - oy*dx;

        float* row = net + m*RS;
        if (hi == 0) {
#pragma unroll
            for (int c = 0; c < 6; ++c) row[c] = pl[c];
            row[126] = 0.f; row[127] = 0.f;            // zero padding to 128
            float f = 1.f;
            for (int fr = 0; fr < 10; ++fr) {
#pragma unroll
                for (int c = 0; c < 6; ++c) row[6 + fr*6 + c] = sinf(pl[c]*f);
                f *= 2.f;
            }
        } else {
            float f = 1.f;
            for (int fr = 0; fr < 10; ++fr) {
#pragma unroll
                for (int c = 0; c < 6; ++c) row[6 + (10+fr)*6 + c] = cosf(pl[c]*f);
                f *= 2.f;
            }
        }
    }
    __syncthreads();

    const float* WtBase = wsW + k*WSLAB;
    const float* bBase  = wsB + k*8*WDIM;

    layer_step<NPAD0>(net, m, hi, lane, WtBase, bBase, ln_s0, ln_b0);
    for (int l = 1; l < 8; ++l) {
        layer_step<WDIM>(net, m, hi, lane,
                         WtBase + NPAD0*WDIM + (l-1)*WDIM*WDIM,
                         bBase + l*WDIM,
                         ln_s + (l-1)*WDIM, ln_b + (l-1)*WDIM);
    }

    // ---- rgb head: 64 -> 3, sigmoid ----
    if (hi == 0) {
        float* row = net + m*RS;
        float o0 = rgbB[0], o1 = rgbB[1], o2 = rgbB[2];
        for (int n = 0; n < WDIM; ++n) {
            float x = row[n];
            o0 = fmaf(x, rgbW[n*3 + 0], o0);
            o1 = fmaf(x, rgbW[n*3 + 1], o1);
            o2 = fmaf(x, rgbW[n*3 + 2], o2);
        }
        float* op = out + gray*3;
        op[0] = 1.f / (1.f + expf(-o0));
        op[1] = 1.f / (1.f + expf(-o1));
        op[2] = 1.f / (1.f + expf(-o2));
    }
}

// ---------------------------------------------------------------------------
extern "C" void kernel_launch(void* const* d_in, const int* in_sizes, int n_in,
                              void* d_out, int out_size, void* d_ws, size_t ws_size,
                              hipStream_t stream)
{
    (void)in_sizes; (void)n_in; (void)out_size; (void)ws_size;

    const float* rays = (const float*)d_in[0];
    const float* lat  = (const float*)d_in[1];
    const float* temb = (const float*)d_in[2];
    const float* hW0  = (const float*)d_in[3];
    const float* hb0  = (const float*)d_in[4];
    const float* hls0 = (const float*)d_in[5];
    const float* hlb0 = (const float*)d_in[6];
    const float* wW0  = (const float*)d_in[7];
    const float* wb0  = (const float*)d_in[8];
    const float* bbW0 = (const float*)d_in[9];
    const float* bbb0 = (const float*)d_in[10];
    const float* lns0 = (const float*)d_in[11];
    const float* lnb0 = (const float*)d_in[12];
    const float* hW   = (const float*)d_in[13];
    const float* hb   = (const float*)d_in[14];
    const float* hls  = (const float*)d_in[15];
    const float* hlb  = (const float*)d_in[16];
    const float* wW   = (const float*)d_in[17];
    const float* wb   = (const float*)d_in[18];
    const float* bbW  = (const float*)d_in[19];
    const float* bbb  = (const float*)d_in[20];
    const float* lns  = (const float*)d_in[21];
    const float* lnb  = (const float*)d_in[22];
    const float* rgbW = (const float*)d_in[23];
    const float* rgbB = (const float*)d_in[24];

    float* outp = (float*)d_out;
    float* wsW  = (float*)d_ws;
    float* wsB  = wsW + WS_B_OFF;      // needs (73728 + 1024)*4 = ~300 KB of ws

    hipLaunchKernelGGL(hyper_weights, dim3(16), dim3(256), 0, stream,
                       lat, temb, hW0, hb0, hls0, hlb0, wW0, wb0, bbW0, bbb0,
                       hW, hb, hls, hlb, wW, wb, bbW, bbb, wsW, wsB);

    hipLaunchKernelGGL(mlp_forward, dim3((KDIM*PDIM)/64), dim3(128), 0, stream,
                       rays, wsW, wsB, lns0, lnb0, lns, lnb, rgbW, rgbB, outp);
}